// AxisInteractionBlock_30288109371498
// MI455X (gfx1250) — compile-verified
//
#include <hip/hip_runtime.h>

typedef __attribute__((ext_vector_type(2))) float v2f;
typedef __attribute__((ext_vector_type(4))) float v4f;
typedef __attribute__((ext_vector_type(8))) float v8f;

#define D_          256
#define QKV_N       768
#define POS_PER_WG  16
#define ROWS        48          // POS_PER_WG * 3
#define H_STRIDE    260         // 256+4 : stride % 64 == 4 -> conflict-free A feeds
#define QKV_STRIDE  772         // 768+4 : same property
#define QKV_OFF     (ROWS * H_STRIDE)                // 12480 floats
#define S_OFF       (QKV_OFF + ROWS * QKV_STRIDE)    // +37056 floats
#define SMEM_FLOATS (S_OFF + POS_PER_WG * 9)         // +144  => 49680 floats (~194 KB)

__device__ __forceinline__ v8f wmma_f32(v2f a, v2f b, v8f c) {
    return __builtin_amdgcn_wmma_f32_16x16x4_f32(false, a, false, b, (short)0, c,
                                                 false, false);
}

__global__ __launch_bounds__(256, 1)
void fused_axis_block(const float* __restrict__ x,
                      const float* __restrict__ gamma,
                      const float* __restrict__ beta,
                      const float* __restrict__ Wqkv,
                      const float* __restrict__ bqkv,
                      const float* __restrict__ Wproj,
                      const float* __restrict__ bproj,
                      const float* __restrict__ Wfuse,
                      const float* __restrict__ bfuse,
                      float* __restrict__ out)
{
    __shared__ float smem[SMEM_FLOATS];
    float* hbuf = smem;                 // [48][260] : h, later attn-out
    float* qkv  = smem + QKV_OFF;       // [48][772] : qkv, later yfuse[16][772]
    float* sbuf = smem + S_OFF;         // [16][3][3] scores / attn weights

    const int tid   = threadIdx.x;
    const int wave  = tid >> 5;
    const int lane  = tid & 31;
    const int nlane = lane & 15;
    const int khalf = (lane >> 4) * 2;      // K offset 0 or 2 for WMMA A/B frags
    const int rhalf = (lane >> 4) * 8;      // M offset 0 or 8 for WMMA C/D rows
    const size_t rowbase = (size_t)blockIdx.x * ROWS;

    // ---------- Phase 0: LayerNorm: x -> hbuf (wave handles 6 rows) ----------
    #pragma unroll
    for (int i = 0; i < 6; ++i) {
        const int r = wave * 6 + i;
        const float* xr = x + (rowbase + r) * D_;
        v4f a = *(const v4f*)(xr + lane * 8);
        v4f b = *(const v4f*)(xr + lane * 8 + 4);
        float s  = a.x + a.y + a.z + a.w + b.x + b.y + b.z + b.w;
        float s2 = a.x*a.x + a.y*a.y + a.z*a.z + a.w*a.w
                 + b.x*b.x + b.y*b.y + b.z*b.z + b.w*b.w;
        #pragma unroll
        for (int off = 16; off > 0; off >>= 1) {
            s  += __shfl_xor(s,  off, 32);
            s2 += __shfl_xor(s2, off, 32);
        }
        const float mu  = s  * (1.0f / D_);
        const float var = s2 * (1.0f / D_) - mu * mu;
        const float rs  = rsqrtf(var + 1e-5f);
        v4f g0 = *(const v4f*)(gamma + lane * 8);
        v4f g1 = *(const v4f*)(gamma + lane * 8 + 4);
        v4f b0 = *(const v4f*)(beta  + lane * 8);
        v4f b1 = *(const v4f*)(beta  + lane * 8 + 4);
        v4f h0, h1;
        #pragma unroll
        for (int j = 0; j < 4; ++j) {
            h0[j] = (a[j] - mu) * rs * g0[j] + b0[j];
            h1[j] = (b[j] - mu) * rs * g1[j] + b1[j];
        }
        *(v4f*)(hbuf + r * H_STRIDE + lane * 8)     = h0;
        *(v4f*)(hbuf + r * H_STRIDE + lane * 8 + 4) = h1;
    }
    __syncthreads();

    // ---------- Phase 1: QKV GEMM  (48x256 @ 256x768) -> qkv in LDS ----------
    for (int nt = wave; nt < QKV_N / 16; nt += 8) {
        const int n = nt * 16 + nlane;
        v8f acc0 = {}, acc1 = {}, acc2 = {};
        #pragma unroll 4
        for (int k = 0; k < D_; k += 4) {
            const int kk = k + khalf;
            v2f bf;
            bf.x = Wqkv[(size_t)kk * QKV_N + n];
            bf.y = Wqkv[(size_t)(kk + 1) * QKV_N + n];
            v2f a0 = *(const v2f*)(hbuf + (nlane)      * H_STRIDE + kk);
            v2f a1 = *(const v2f*)(hbuf + (16 + nlane) * H_STRIDE + kk);
            v2f a2 = *(const v2f*)(hbuf + (32 + nlane) * H_STRIDE + kk);
            acc0 = wmma_f32(a0, bf, acc0);
            acc1 = wmma_f32(a1, bf, acc1);
            acc2 = wmma_f32(a2, bf, acc2);
        }
        const float bn = bqkv[n];
        #pragma unroll
        for (int j = 0; j < 8; ++j) {
            qkv[(rhalf + j)      * QKV_STRIDE + n] = acc0[j] + bn;
            qkv[(16 + rhalf + j) * QKV_STRIDE + n] = acc1[j] + bn;
            qkv[(32 + rhalf + j) * QKV_STRIDE + n] = acc2[j] + bn;
        }
    }
    __syncthreads();

    // ---------- Phase 2: 3x3 attention scores (q.k / sqrt(256)) ----------
    if (tid < POS_PER_WG * 9) {
        const int pos = tid / 9;
        const int ci  = (tid % 9) / 3;
        const int cj  = tid % 3;
        const float* qrow = qkv + (pos * 3 + ci) * QKV_STRIDE;         // q
        const float* krow = qkv + (pos * 3 + cj) * QKV_STRIDE + 256;   // k
        float s = 0.0f;
        #pragma unroll 4
        for (int d = 0; d < D_; d += 4) {
            v4f qv = *(const v4f*)(qrow + d);
            v4f kv = *(const v4f*)(krow + d);
            s += qv.x*kv.x + qv.y*kv.y + qv.z*kv.z + qv.w*kv.w;
        }
        sbuf[tid] = s * (1.0f / 16.0f);
    }
    __syncthreads();

    // ---------- Phase 3: softmax over cj (3 values) ----------
    if (tid < ROWS) {
        float* sr = sbuf + (tid / 3) * 9 + (tid % 3) * 3;
        const float m  = fmaxf(sr[0], fmaxf(sr[1], sr[2]));
        const float e0 = __expf(sr[0] - m);
        const float e1 = __expf(sr[1] - m);
        const float e2 = __expf(sr[2] - m);
        const float inv = 1.0f / (e0 + e1 + e2);
        sr[0] = e0 * inv; sr[1] = e1 * inv; sr[2] = e2 * inv;
    }
    __syncthreads();

    // ---------- Phase 4: attn @ v -> hbuf (h is dead now) ----------
    {
        const int d = tid;   // 0..255, one output column per thread
        for (int row = 0; row < ROWS; ++row) {
            const int pos = row / 3;
            const float* ar = sbuf + pos * 9 + (row - pos * 3) * 3;
            const float* v0 = qkv + (pos * 3) * QKV_STRIDE + 512 + d;
            float o = ar[0] * v0[0]
                    + ar[1] * v0[QKV_STRIDE]
                    + ar[2] * v0[2 * QKV_STRIDE];
            hbuf[row * H_STRIDE + d] = o;
        }
    }
    __syncthreads();

    // ---------- Phase 5: proj GEMM (48x256 @ 256x256) -> yfuse[16][772] ----------
    // yfuse aliases the qkv region; qkv (v) is dead after phase 4's barrier.
    for (int nt = wave; nt < D_ / 16; nt += 8) {
        const int n = nt * 16 + nlane;
        v8f acc0 = {}, acc1 = {}, acc2 = {};
        #pragma unroll 4
        for (int k = 0; k < D_; k += 4) {
            const int kk = k + khalf;
            v2f bf;
            bf.x = Wproj[(size_t)kk * D_ + n];
            bf.y = Wproj[(size_t)(kk + 1) * D_ + n];
            v2f a0 = *(const v2f*)(hbuf + (nlane)      * H_STRIDE + kk);
            v2f a1 = *(const v2f*)(hbuf + (16 + nlane) * H_STRIDE + kk);
            v2f a2 = *(const v2f*)(hbuf + (32 + nlane) * H_STRIDE + kk);
            acc0 = wmma_f32(a0, bf, acc0);
            acc1 = wmma_f32(a1, bf, acc1);
            acc2 = wmma_f32(a2, bf, acc2);
        }
        const float bn = bproj[n];
        #pragma unroll
        for (int j = 0; j < 8; ++j) {
            int row, pos, c;
            row = rhalf + j;      pos = row / 3; c = row - pos * 3;
            qkv[pos * QKV_STRIDE + c * 256 + n] = acc0[j] + bn;
            row = 16 + rhalf + j; pos = row / 3; c = row - pos * 3;
            qkv[pos * QKV_STRIDE + c * 256 + n] = acc1[j] + bn;
            row = 32 + rhalf + j; pos = row / 3; c = row - pos * 3;
            qkv[pos * QKV_STRIDE + c * 256 + n] = acc2[j] + bn;
        }
    }
    __syncthreads();

    // ---------- Phase 6: fuse GEMM (16x768 @ 768x256) + bias -> global out ----------
    for (int nt = wave; nt < D_ / 16; nt += 8) {
        const int n = nt * 16 + nlane;
        v8f acc = {};
        #pragma unroll 4
        for (int k = 0; k < QKV_N; k += 4) {
            const int kk = k + khalf;
            v2f bf;
            bf.x = Wfuse[(size_t)kk * D_ + n];
            bf.y = Wfuse[(size_t)(kk + 1) * D_ + n];
            v2f a = *(const v2f*)(qkv + nlane * QKV_STRIDE + kk);
            acc = wmma_f32(a, bf, acc);
        }
        const float bn = bfuse[n];
        #pragma unroll
        for (int j = 0; j < 8; ++j) {
            const int pos = rhalf + j;
            out[((size_t)blockIdx.x * POS_PER_WG + pos) * D_ + n] = acc[j] + bn;
        }
    }
}

extern "C" void kernel_launch(void* const* d_in, const int* in_sizes, int n_in,
                              void* d_out, int out_size, void* d_ws, size_t ws_size,
                              hipStream_t stream) {
    (void)n_in; (void)d_ws; (void)ws_size; (void)out_size;
    const float* x     = (const float*)d_in[0];
    const float* gamma = (const float*)d_in[1];
    const float* beta  = (const float*)d_in[2];
    const float* Wqkv  = (const float*)d_in[3];
    const float* bqkv  = (const float*)d_in[4];
    const float* Wproj = (const float*)d_in[5];
    const float* bproj = (const float*)d_in[6];
    const float* Wfuse = (const float*)d_in[7];
    const float* bfuse = (const float*)d_in[8];
    float* out = (float*)d_out;

    const int Dd   = in_sizes[1];              // 256
    const int rows = in_sizes[0] / Dd;         // B*NP*C = 196608
    const int npos = rows / 3;                 // B*NP   = 65536
    const int grid = npos / POS_PER_WG;        // 4096 workgroups

    fused_axis_block<<<grid, 256, 0, stream>>>(x, gamma, beta, Wqkv, bqkv,
                                               Wproj, bproj, Wfuse, bfuse, out);
}